// MutualConsistency_51316269253469
// MI455X (gfx1250) — compile-verified
//
#include <hip/hip_runtime.h>

typedef __attribute__((ext_vector_type(2))) float v2f;
typedef __attribute__((ext_vector_type(8))) float v8f;

#define MASK_N (64 * 512 * 512)   // elements per mask
#define CB 64                     // batches
#define CN 512                    // contour points
#define CF 1024                   // flattened contour length (N * 2 channels)

// ws layout (floats): 0:sum(c*g) 1:sum(c) 2:sum(g) 3:sum(c*s) 4:sum(s) 5:seg_mean 6:cons_mean
__global__ void init_ws_kernel(float* __restrict__ ws) {
    if (threadIdx.x < 8) ws[threadIdx.x] = 0.0f;
}

// ---------------- HBM-bound mask reductions (dominant cost) ----------------
__global__ __launch_bounds__(256) void mask_reduce_kernel(
        const float4* __restrict__ gt, const float4* __restrict__ sn,
        const float4* __restrict__ cl, float* __restrict__ ws) {
    const int n4 = MASK_N / 4;
    int tid = blockIdx.x * blockDim.x + threadIdx.x;
    int stride = gridDim.x * blockDim.x;
    float a_cg = 0.f, a_c = 0.f, a_g = 0.f, a_cs = 0.f, a_s = 0.f;
    for (int i = tid; i < n4; i += stride) {
        float4 g = gt[i], s = sn[i], c = cl[i];
        a_cg += c.x * g.x + c.y * g.y + c.z * g.z + c.w * g.w;
        a_cs += c.x * s.x + c.y * s.y + c.z * s.z + c.w * s.w;
        a_c  += c.x + c.y + c.z + c.w;
        a_g  += g.x + g.y + g.z + g.w;
        a_s  += s.x + s.y + s.z + s.w;
    }
    // wave32 reduction
    for (int off = 16; off; off >>= 1) {
        a_cg += __shfl_xor(a_cg, off, 32);
        a_c  += __shfl_xor(a_c,  off, 32);
        a_g  += __shfl_xor(a_g,  off, 32);
        a_cs += __shfl_xor(a_cs, off, 32);
        a_s  += __shfl_xor(a_s,  off, 32);
    }
    __shared__ float red[8][5];
    int wave = threadIdx.x >> 5;
    if ((threadIdx.x & 31) == 0) {
        red[wave][0] = a_cg; red[wave][1] = a_c; red[wave][2] = a_g;
        red[wave][3] = a_cs; red[wave][4] = a_s;
    }
    __syncthreads();
    if (threadIdx.x == 0) {
        float t0 = 0.f, t1 = 0.f, t2 = 0.f, t3 = 0.f, t4 = 0.f;
        for (int w = 0; w < 8; ++w) {
            t0 += red[w][0]; t1 += red[w][1]; t2 += red[w][2];
            t3 += red[w][3]; t4 += red[w][4];
        }
        atomicAdd(&ws[0], t0); atomicAdd(&ws[1], t1); atomicAdd(&ws[2], t2);
        atomicAdd(&ws[3], t3); atomicAdd(&ws[4], t4);
    }
}

// -------- min-shift MSE via V_WMMA_F32_16X16X4_F32 (Hankel x Toeplitz) --------
// one block of 64 threads (2 waves) per (batch, pair); wave w covers so in [16w, 16w+16)
__global__ __launch_bounds__(64) void contour_minmse_kernel(
        const float* __restrict__ gt_contour,    // ref,  pair 0
        const float* __restrict__ snake_gt,      // pred, pair 0
        const float* __restrict__ snake_classic, // pred, pair 1
        const float* __restrict__ classic_ct,    // ref,  pair 1
        float* __restrict__ ws) {
    __shared__ float pL[CF];
    __shared__ float rL[CF];
    __shared__ float sred[64];

    const int b    = blockIdx.x >> 1;
    const int pair = blockIdx.x & 1;
    const float* __restrict__ pb = (pair ? snake_classic : snake_gt) + b * CF;
    const float* __restrict__ rb = (pair ? classic_ct    : gt_contour) + b * CF;

    const int t = threadIdx.x;
    float p2 = 0.f, r2 = 0.f;
    for (int i = t; i < CF; i += 64) {
        float pv = pb[i], rv = rb[i];
        pL[i] = pv; rL[i] = rv;
        p2 += pv * pv; r2 += rv * rv;
    }
    sred[t] = p2 + r2;
    __syncthreads();
    for (int off = 32; off; off >>= 1) {
        if (t < off) sred[t] += sred[t + off];
        __syncthreads();
    }
    const float sumsq = sred[0];
    __syncthreads();

    const int lane = t & 31;
    const int wave = t >> 5;
    const int M  = lane & 15;             // A row (si) / B column (so_local)
    const int K0 = (lane >> 4) << 1;      // this lane's K slots: {K0, K0+1}
    const int aShift = 2 * M;             // A[si][k]  = pL[(k + 2*si)  mod CF]
    const int bShift = 32 * ((wave << 4) + M); // B[k][so] = rL[(k - 32*so) mod CF]

    v8f acc = {0.f, 0.f, 0.f, 0.f, 0.f, 0.f, 0.f, 0.f};
    for (int k0 = 0; k0 < CF; k0 += 4) {
        const int ka = k0 + K0;
        v2f a, bf;
        a.x  = pL[(ka     + aShift) & (CF - 1)];
        a.y  = pL[(ka + 1 + aShift) & (CF - 1)];
        bf.x = rL[(ka     - bShift + 4 * CF) & (CF - 1)];
        bf.y = rL[(ka + 1 - bShift + 4 * CF) & (CF - 1)];
        acc = __builtin_amdgcn_wmma_f32_16x16x4_f32(
            /*neg_a=*/false, a, /*neg_b=*/false, bf,
            /*c_mod=*/(short)0, acc, /*reuse_a=*/false, /*reuse_b=*/false);
    }

    // acc[v] at lane l holds c[s], s = (v + (l<16?0:8)) + 16*(16*wave + (l&15));
    // the 64 threads x 8 VGPRs tile exactly the 512 shifts, so block-min == min over s.
    float mn = 3.402823e38f;
    #pragma unroll
    for (int v = 0; v < 8; ++v) {
        float mse = (sumsq - 2.0f * acc[v]) * (1.0f / (float)CF);
        mn = fminf(mn, mse);
    }
    __syncthreads();
    sred[t] = mn;
    __syncthreads();
    for (int off = 32; off; off >>= 1) {
        if (t < off) sred[t] = fminf(sred[t], sred[t + off]);
        __syncthreads();
    }
    if (t == 0) atomicAdd(&ws[5 + pair], sred[0] * (1.0f / (float)CB));
}

// ---------------------------- scalar combine ----------------------------
__global__ void finalize_kernel(const float* __restrict__ ws, float* __restrict__ out) {
    if (threadIdx.x == 0 && blockIdx.x == 0) {
        const float s_cg = ws[0], s_c = ws[1], s_g = ws[2], s_cs = ws[3], s_s = ws[4];
        const float seg = ws[5], cons = ws[6];
        const float dice1 = 1.0f - (2.0f * s_cg + 1.0f) / (s_c + s_g + 1.0f);
        const float dice2 = 1.0f - (2.0f * s_cs + 1.0f) / (s_c + s_s + 1.0f);
        out[0] = 0.5f * (dice1 + seg) + 0.5f * (dice2 + cons); // GAMMA = 0.5
    }
}

extern "C" void kernel_launch(void* const* d_in, const int* in_sizes, int n_in,
                              void* d_out, int out_size, void* d_ws, size_t ws_size,
                              hipStream_t stream) {
    const float* gt_mask  = (const float*)d_in[0];
    const float* gt_ct    = (const float*)d_in[1];
    const float* snk_gt   = (const float*)d_in[2];
    const float* snk_cl   = (const float*)d_in[3];
    const float* snk_mask = (const float*)d_in[4];
    const float* cl_ct    = (const float*)d_in[5];
    const float* cl_mask  = (const float*)d_in[6];
    float* ws  = (float*)d_ws;
    float* out = (float*)d_out;

    hipLaunchKernelGGL(init_ws_kernel, dim3(1), dim3(32), 0, stream, ws);
    hipLaunchKernelGGL(mask_reduce_kernel, dim3(4096), dim3(256), 0, stream,
                       (const float4*)gt_mask, (const float4*)snk_mask,
                       (const float4*)cl_mask, ws);
    hipLaunchKernelGGL(contour_minmse_kernel, dim3(CB * 2), dim3(64), 0, stream,
                       gt_ct, snk_gt, snk_cl, cl_ct, ws);
    hipLaunchKernelGGL(finalize_kernel, dim3(1), dim3(32), 0, stream, ws, out);
}